// SE3MultiSelfAttention_90254442758484
// MI455X (gfx1250) — compile-verified
//
#include <hip/hip_runtime.h>
#include <cmath>

#define BB 2
#define HH 8
#define NN 1000
#define NP 1024
#define DIMD 256
#define QDQ 32
#define K288 288

typedef float v2f __attribute__((ext_vector_type(2)));
typedef float v8f __attribute__((ext_vector_type(8)));
typedef int v4i __attribute__((ext_vector_type(4)));

#if __has_builtin(__builtin_amdgcn_global_load_async_to_lds_b128)
#define SE3_ASYNC 1
#endif

__device__ __forceinline__ v8f wmma4(v2f a, v2f b, v8f c) {
  // D = A(16x4 f32) * B(4x16 f32) + C(16x16 f32); V_WMMA_F32_16X16X4_F32
  return __builtin_amdgcn_wmma_f32_16x16x4_f32(false, a, false, b, (short)0, c,
                                               false, false);
}

__device__ __forceinline__ v2f mkv2(float a, float b) {
  v2f t; t.x = a; t.y = b; return t;
}

// 16B global -> LDS copy: async DMA path on gfx1250 (ASYNCcnt), else ld+st.
__device__ __forceinline__ void cp16(const float* g, float* l) {
#ifdef SE3_ASYNC
  __builtin_amdgcn_global_load_async_to_lds_b128(
      (__attribute__((address_space(1))) v4i*)g,
      (__attribute__((address_space(3))) v4i*)l, 0, 0);
#else
  float4 v = *(const float4*)g;
  *(float4*)l = v;
#endif
}

template <int Ncnt>
__device__ __forceinline__ void awaitcnt() {
#ifdef SE3_ASYNC
#if __has_builtin(__builtin_amdgcn_s_wait_asynccnt)
  __builtin_amdgcn_s_wait_asynccnt(Ncnt);
#else
  asm volatile("s_wait_asynccnt %0" ::"i"(Ncnt) : "memory");
#endif
#endif
}

// ---------------------------------------------------------------------------
__global__ void se3_zero(float4* __restrict__ p, size_t n4) {
  size_t i = (size_t)blockIdx.x * blockDim.x + threadIdx.x;
  size_t stride = (size_t)gridDim.x * blockDim.x;
  float4 z; z.x = 0.f; z.y = 0.f; z.z = 0.f; z.w = 0.f;
  for (; i < n4; i += stride) p[i] = z;
}

// ---------------------------------------------------------------------------
// Embedding projection: E[t*256+d, n] = sum_e emb[t,d,e] * x[e,n]  (complex)
__global__ void se3_emb(const float2* __restrict__ emb, const float2* __restrict__ x,
                        int m_l, int kbase,
                        float* __restrict__ Qr, float* __restrict__ Qi,
                        float* __restrict__ Kr, float* __restrict__ Ki,
                        float* __restrict__ Brr, float* __restrict__ Bii) {
  const int lane = threadIdx.x & 31;
  const int half = lane >> 4;
  const int l16 = lane & 15;
  const int Mbase = blockIdx.x * 32;
  const int Nbase = (blockIdx.y * 4 + threadIdx.y) * 32;
  const int bm = blockIdx.z;
  const int b = bm / m_l, mm = bm % m_l;
  const float2* xb = x + (size_t)bm * DIMD * NN;

  v8f aR[2][2] = {}, aI[2][2] = {};
#pragma unroll 1
  for (int k0 = 0; k0 < DIMD; k0 += 4) {
    const int e0 = k0 + 2 * half, e1 = e0 + 1;
    if (k0 + 4 < DIMD) {
      __builtin_prefetch(emb + (size_t)(Mbase + l16) * DIMD + e0 + 4, 0, 1);
      __builtin_prefetch(xb + (size_t)(e0 + 4) * NN + Nbase, 0, 1);
    }
    v2f arR[2], arI[2], anI[2];
#pragma unroll
    for (int i = 0; i < 2; ++i) {
      int row = Mbase + 16 * i + l16;
      float2 w0 = emb[(size_t)row * DIMD + e0];
      float2 w1 = emb[(size_t)row * DIMD + e1];
      arR[i] = mkv2(w0.x, w1.x);
      arI[i] = mkv2(w0.y, w1.y);
      anI[i] = -arI[i];
    }
    v2f brR[2], brI[2];
#pragma unroll
    for (int j = 0; j < 2; ++j) {
      int col = Nbase + 16 * j + l16;
      float2 x0, x1;
      x0.x = 0.f; x0.y = 0.f; x1.x = 0.f; x1.y = 0.f;
      if (col < NN) {
        x0 = xb[(size_t)e0 * NN + col];
        x1 = xb[(size_t)e1 * NN + col];
      }
      brR[j] = mkv2(x0.x, x1.x);
      brI[j] = mkv2(x0.y, x1.y);
    }
#pragma unroll
    for (int i = 0; i < 2; ++i)
#pragma unroll
      for (int j = 0; j < 2; ++j) {
        aR[i][j] = wmma4(arR[i], brR[j], aR[i][j]);
        aR[i][j] = wmma4(anI[i], brI[j], aR[i][j]);
        aI[i][j] = wmma4(arR[i], brI[j], aI[i][j]);
        aI[i][j] = wmma4(arI[i], brR[j], aI[i][j]);
      }
  }
#pragma unroll
  for (int i = 0; i < 2; ++i)
#pragma unroll
    for (int j = 0; j < 2; ++j)
#pragma unroll
      for (int v = 0; v < 8; ++v) {
        int row = Mbase + 16 * i + v + 8 * half;
        int col = Nbase + 16 * j + l16;
        int t = row >> 8, d = row & 255;
        int h = d >> 5, c = d & 31;
        size_t idx = ((size_t)(b * HH + h) * K288 + kbase + mm * QDQ + c) * NP + col;
        float re = aR[i][j][v], im = aI[i][j][v];
        if (t == 0)      { Qr[idx] = re; Qi[idx] = -im; }   // conj(query)
        else if (t == 1) { Kr[idx] = re; Ki[idx] = im; }
        else             { Brr[idx] = re; Bii[idx] = im; }
      }
}

// ---------------------------------------------------------------------------
// T[bh, mi, n] = sum_c conj(Q)[bh, kb+mm*32+c, n] * enc_l[h, c]
__global__ void se3_posT(const float2* __restrict__ enc0, const float2* __restrict__ enc1,
                         const float2* __restrict__ enc2,
                         const float* __restrict__ Qr, const float* __restrict__ Qi,
                         float* __restrict__ Tr, float* __restrict__ Ti) {
  size_t tid = (size_t)blockIdx.x * blockDim.x + threadIdx.x;
  size_t total = (size_t)BB * HH * 9 * NP;
  if (tid >= total) return;
  int n = (int)(tid % NP);
  int mi = (int)((tid / NP) % 9);
  int bh = (int)(tid / ((size_t)NP * 9));
  int h = bh % HH;
  int kb, mm; const float2* enc;
  if (mi < 1)      { kb = 0;   mm = mi;     enc = enc0; }
  else if (mi < 4) { kb = 32;  mm = mi - 1; enc = enc1; }
  else             { kb = 128; mm = mi - 4; enc = enc2; }
  float sr = 0.f, si = 0.f;
  for (int c = 0; c < QDQ; ++c) {
    size_t qi = ((size_t)bh * K288 + kb + mm * QDQ + c) * NP + n;
    float qr = Qr[qi], qm = Qi[qi];
    float2 e = enc[h * QDQ + c];
    sr += qr * e.x - qm * e.y;
    si += qr * e.y + qm * e.x;
  }
  Tr[tid] = sr;
  Ti[tid] = si;
}

// ---------------------------------------------------------------------------
// A[bh,n,j] += s * ( sum_k conjQ[k,n]*K[k,j] + sum_m T[n,m]*Y[n,m,j] )
// Double-buffered async global->LDS staging: 8 K-rows per stage,
// Q slab (shared by the 4 waves) + K slab of 128 j columns.
__global__ void se3_scores(const float* __restrict__ Qr, const float* __restrict__ Qi,
                           const float* __restrict__ Kr, const float* __restrict__ Ki,
                           const float* __restrict__ Tr, const float* __restrict__ Ti,
                           const float2* __restrict__ Y,
                           float* __restrict__ Ar, float* __restrict__ Ai,
                           int m_l, int kbase, int mqd, int tbase) {
  const int lane = threadIdx.x & 31;
  const int half = lane >> 4;
  const int l16 = lane & 15;
  const int wj = threadIdx.y;               // wave's 32-wide j subtile
  const int tid = wj * 32 + threadIdx.x;    // 0..127
  const int n0 = blockIdx.x * 32;
  const int jblk = blockIdx.y * 128;
  const int bh = blockIdx.z;
  const size_t qrow0 = (size_t)bh * K288 + kbase;

  __shared__ float sQ[2][2][8][32];    // [buf][r/i][krow][n]   4 KB
  __shared__ float sK[2][2][8][128];   // [buf][r/i][krow][j]  16 KB

  auto issue = [&](int k0, int buf) {
    {  // Q slab: 128 x b128 chunks, 1 per thread (1 async instr / wave)
      int c = tid;
      int p = c >> 6, row = (c >> 3) & 7, seg = c & 7;
      const float* g = (p ? Qi : Qr) + (qrow0 + k0 + row) * NP + n0 + seg * 4;
      cp16(g, &sQ[buf][p][row][seg * 4]);
    }
#pragma unroll
    for (int q = 0; q < 4; ++q) {  // K slab: 512 chunks (4 async instr / wave)
      int c = tid + q * 128;
      int p = c >> 8, row = (c >> 5) & 7, seg = c & 31;
      const float* g = (p ? Ki : Kr) + (qrow0 + k0 + row) * NP + jblk + seg * 4;
      cp16(g, &sK[buf][p][row][seg * 4]);
    }
  };

  const int nst = mqd / 8;
  v8f aRf[2][2] = {}, aIf[2][2] = {};
  issue(0, 0);
#pragma unroll 1
  for (int s = 0; s < nst; ++s) {
    if (s + 1 < nst) {
      issue((s + 1) * 8, (s + 1) & 1);
      awaitcnt<5>();      // stage s copies done; next stage's 5 stay in flight
    } else {
      awaitcnt<0>();
    }
    __syncthreads();
    const int buf = s & 1;
#pragma unroll
    for (int kk = 0; kk < 8; kk += 4) {
      const int r0 = kk + 2 * half, r1 = r0 + 1;
      v2f qR[2], qI[2], qnI[2];
#pragma unroll
      for (int i = 0; i < 2; ++i) {
        int n = 16 * i + l16;
        qR[i] = mkv2(sQ[buf][0][r0][n], sQ[buf][0][r1][n]);
        qI[i] = mkv2(sQ[buf][1][r0][n], sQ[buf][1][r1][n]);
        qnI[i] = -qI[i];
      }
      v2f kR[2], kI[2];
#pragma unroll
      for (int j = 0; j < 2; ++j) {
        int jc = wj * 32 + 16 * j + l16;
        kR[j] = mkv2(sK[buf][0][r0][jc], sK[buf][0][r1][jc]);
        kI[j] = mkv2(sK[buf][1][r0][jc], sK[buf][1][r1][jc]);
      }
#pragma unroll
      for (int i = 0; i < 2; ++i)
#pragma unroll
        for (int j = 0; j < 2; ++j) {
          aRf[i][j] = wmma4(qR[i], kR[j], aRf[i][j]);
          aRf[i][j] = wmma4(qnI[i], kI[j], aRf[i][j]);
          aIf[i][j] = wmma4(qR[i], kI[j], aIf[i][j]);
          aIf[i][j] = wmma4(qI[i], kR[j], aIf[i][j]);
        }
    }
    __syncthreads();
  }

  const float s5 = 0.17677669529663688f;  // 1/sqrt(32)
#pragma unroll
  for (int i = 0; i < 2; ++i)
#pragma unroll
    for (int j = 0; j < 2; ++j)
#pragma unroll
      for (int v = 0; v < 8; ++v) {
        int n = n0 + 16 * i + v + 8 * half;
        int jc = jblk + wj * 32 + 16 * j + l16;
        float pr = 0.f, pi = 0.f;
        if (n < NN && jc < NN) {
          for (int mmi = 0; mmi < m_l; ++mmi) {
            size_t ti = ((size_t)bh * 9 + tbase + mmi) * NP + n;
            float tr = Tr[ti], tii = Ti[ti];
            float2 y = Y[((size_t)n * m_l + mmi) * NN + jc];
            pr += tr * y.x - tii * y.y;
            pi += tr * y.y + tii * y.x;
          }
        }
        size_t idx = ((size_t)bh * NP + n) * NP + jc;
        Ar[idx] += s5 * (aRf[i][j][v] + pr);
        Ai[idx] += s5 * (aIf[i][j][v] + pi);
      }
}

// ---------------------------------------------------------------------------
// Row softmax over |A|; real weights, zero-pads cols [NN, NP).
__global__ void se3_softmax(const float* __restrict__ Ar, const float* __restrict__ Ai,
                            float* __restrict__ Aw) {
  __shared__ float red[256];
  const int row = blockIdx.x;
  const int bh = row / NN, n = row % NN;
  const size_t base = ((size_t)bh * NP + n) * NP;
  const int t = threadIdx.x;
  float mx = -1e30f;
  for (int j = t; j < NN; j += 256) {
    float r = Ar[base + j], im = Ai[base + j];
    mx = fmaxf(mx, sqrtf(r * r + im * im));
  }
  red[t] = mx; __syncthreads();
  for (int sft = 128; sft > 0; sft >>= 1) {
    if (t < sft) red[t] = fmaxf(red[t], red[t + sft]);
    __syncthreads();
  }
  mx = red[0]; __syncthreads();
  float sum = 0.f;
  for (int j = t; j < NN; j += 256) {
    float r = Ar[base + j], im = Ai[base + j];
    float v = __expf(sqrtf(r * r + im * im) - mx);
    Aw[base + j] = v;
    sum += v;
  }
  red[t] = sum; __syncthreads();
  for (int sft = 128; sft > 0; sft >>= 1) {
    if (t < sft) red[t] += red[t + sft];
    __syncthreads();
  }
  float inv = 1.f / red[0];
  for (int j = t; j < NN; j += 256) Aw[base + j] *= inv;
  for (int j = NN + t; j < NP; j += 256) Aw[base + j] = 0.f;
}

// ---------------------------------------------------------------------------
// V[bh,k,n] = sum_j Bv[bh,k,j] * Aw[bh,n,j]  (Aw real -> 2 real GEMMs)
// Async-staged: Bv slab (shared across 4 waves) + Aw slab, 8 j per stage.
__global__ void se3_av(const float* __restrict__ Brr, const float* __restrict__ Bii,
                       const float* __restrict__ Aw,
                       float* __restrict__ Vr, float* __restrict__ Vi, int kbase) {
  const int lane = threadIdx.x & 31;
  const int half = lane >> 4;
  const int l16 = lane & 15;
  const int wj = threadIdx.y;
  const int tid = wj * 32 + threadIdx.x;
  const int Mb = blockIdx.x * 32;
  const int nblk = blockIdx.y * 128;
  const int bh = blockIdx.z;
  const size_t prow = (size_t)bh * K288 + kbase;
  const size_t awb = (size_t)bh * NP * NP;

  __shared__ float sB[2][2][32][8];    // [buf][r/i][k-row][j]  4 KB
  __shared__ float sA[2][128][8];      // [buf][n][j]           8 KB

  auto issue = [&](int j0, int buf) {
    {  // Bv slab: 128 chunks (1 async instr / wave)
      int c = tid;
      int p = c >> 6, rem = c & 63, row = rem >> 1, chk = rem & 1;
      const float* g = (p ? Bii : Brr) + (prow + Mb + row) * NP + j0 + chk * 4;
      cp16(g, &sB[buf][p][row][chk * 4]);
    }
#pragma unroll
    for (int q = 0; q < 2; ++q) {  // Aw slab: 256 chunks (2 async instr / wave)
      int c = tid + q * 128;
      int row = c >> 1, chk = c & 1;
      const float* g = Aw + awb + (size_t)(nblk + row) * NP + j0 + chk * 4;
      cp16(g, &sA[buf][row][chk * 4]);
    }
  };

  v8f vR[2][2] = {}, vI[2][2] = {};
  issue(0, 0);
#pragma unroll 1
  for (int s = 0; s < NP / 8; ++s) {
    if (s + 1 < NP / 8) {
      issue((s + 1) * 8, (s + 1) & 1);
      awaitcnt<3>();
    } else {
      awaitcnt<0>();
    }
    __syncthreads();
    const int buf = s & 1;
#pragma unroll
    for (int kk = 0; kk < 8; kk += 4) {
      const int e0 = kk + 2 * half;
      v2f aR[2], aI2[2];
#pragma unroll
      for (int i = 0; i < 2; ++i) {
        int ml = 16 * i + l16;
        aR[i] = mkv2(sB[buf][0][ml][e0], sB[buf][0][ml][e0 + 1]);
        aI2[i] = mkv2(sB[buf][1][ml][e0], sB[buf][1][ml][e0 + 1]);
      }
      v2f bw[2];
#pragma unroll
      for (int j = 0; j < 2; ++j) {
        int nl = wj * 32 + 16 * j + l16;
        bw[j] = mkv2(sA[buf][nl][e0], sA[buf][nl][e0 + 1]);
      }
#pragma unroll
      for (int i = 0; i < 2; ++i)
#pragma unroll
        for (int j = 0; j < 2; ++j) {
          vR[i][j] = wmma4(aR[i], bw[j], vR[i][j]);
          vI[i][j] = wmma4(aI2[i], bw[j], vI[i][j]);
        }
    }
    __syncthreads();
  }
#pragma unroll
  for (int i = 0; i < 2; ++i)
#pragma unroll
    for (int j = 0; j < 2; ++j)
#pragma unroll
      for (int v = 0; v < 8; ++v) {
        int rowk = Mb + 16 * i + v + 8 * half;
        int col = nblk + wj * 32 + 16 * j + l16;
        size_t idx = (prow + rowk) * NP + col;
        Vr[idx] = vR[i][j][v];
        Vi[idx] = vI[i][j][v];
      }
}

// ---------------------------------------------------------------------------
// Out[b,m,d,n] = sum_e outw[d,e] * Vmerged[b,m,e,n]  (complex)
__global__ void se3_outproj(const float2* __restrict__ outw,
                            const float* __restrict__ Vr, const float* __restrict__ Vi,
                            float2* __restrict__ out, int m_l, int kbase, size_t obase) {
  const int lane = threadIdx.x & 31;
  const int half = lane >> 4;
  const int l16 = lane & 15;
  const int d0 = blockIdx.x * 32;
  const int n0 = (blockIdx.y * 4 + threadIdx.y) * 32;
  const int bm = blockIdx.z;
  const int b = bm / m_l, mm = bm % m_l;

  v8f cR[2][2] = {}, cI[2][2] = {};
#pragma unroll 1
  for (int k0 = 0; k0 < DIMD; k0 += 4) {
    const int e0 = k0 + 2 * half, e1 = e0 + 1;
    if (k0 + 4 < DIMD)
      __builtin_prefetch(outw + (size_t)(d0 + l16) * DIMD + e0 + 4, 0, 1);
    v2f wR[2], wI[2], wnI[2];
#pragma unroll
    for (int i = 0; i < 2; ++i) {
      int row = d0 + 16 * i + l16;
      float2 w0 = outw[(size_t)row * DIMD + e0];
      float2 w1 = outw[(size_t)row * DIMD + e1];
      wR[i] = mkv2(w0.x, w1.x);
      wI[i] = mkv2(w0.y, w1.y);
      wnI[i] = -wI[i];
    }
    size_t vr0 = ((size_t)(b * HH + (e0 >> 5)) * K288 + kbase + mm * QDQ + (e0 & 31)) * NP;
    size_t vr1 = ((size_t)(b * HH + (e1 >> 5)) * K288 + kbase + mm * QDQ + (e1 & 31)) * NP;
    v2f bR[2], bI[2];
#pragma unroll
    for (int j = 0; j < 2; ++j) {
      int col = n0 + 16 * j + l16;
      bR[j] = mkv2(Vr[vr0 + col], Vr[vr1 + col]);
      bI[j] = mkv2(Vi[vr0 + col], Vi[vr1 + col]);
    }
#pragma unroll
    for (int i = 0; i < 2; ++i)
#pragma unroll
      for (int j = 0; j < 2; ++j) {
        cR[i][j] = wmma4(wR[i], bR[j], cR[i][j]);
        cR[i][j] = wmma4(wnI[i], bI[j], cR[i][j]);
        cI[i][j] = wmma4(wR[i], bI[j], cI[i][j]);
        cI[i][j] = wmma4(wI[i], bR[j], cI[i][j]);
      }
  }
#pragma unroll
  for (int i = 0; i < 2; ++i)
#pragma unroll
    for (int j = 0; j < 2; ++j)
#pragma unroll
      for (int v = 0; v < 8; ++v) {
        int row = d0 + 16 * i + v + 8 * half;
        int col = n0 + 16 * j + l16;
        if (col < NN) {
          float2 o; o.x = cR[i][j][v]; o.y = cI[i][j][v];
          out[obase + ((size_t)bm * DIMD + row) * NN + col] = o;
        }
      }
}

// ---------------------------------------------------------------------------
extern "C" void kernel_launch(void* const* d_in, const int* in_sizes, int n_in,
                              void* d_out, int out_size, void* d_ws, size_t ws_size,
                              hipStream_t stream) {
  const float2 *x[3], *emb[3], *enc[3], *outw[3], *pos[3];
  for (int l = 0; l < 3; ++l) {
    x[l]    = (const float2*)d_in[l * 5 + 0];
    emb[l]  = (const float2*)d_in[l * 5 + 1];
    enc[l]  = (const float2*)d_in[l * 5 + 2];
    outw[l] = (const float2*)d_in[l * 5 + 3];
    pos[l]  = (const float2*)d_in[l * 5 + 4];
  }
  float* ws = (float*)d_ws;
  const size_t PL = (size_t)BB * HH * K288 * NP;
  const size_t AS = (size_t)BB * HH * NP * NP;
  const size_t TS = (size_t)BB * HH * 9 * NP;
  float *Qr = ws,       *Qi = Qr + PL, *Kr = Qi + PL, *Ki = Kr + PL;
  float *Brr = Ki + PL, *Bii = Brr + PL;
  float *Ar = Bii + PL, *Ai = Ar + AS, *Aw = Ai + AS;
  float *Tr = Aw + AS,  *Ti = Tr + TS;
  float *Vr = Ti + TS,  *Vi = Vr + PL;

  const int mls[3]  = {1, 3, 5};
  const int kb[3]   = {0, 32, 128};
  const int tb[3]   = {0, 1, 4};
  const int mqd[3]  = {32, 96, 160};
  const size_t ob[3] = {0, 512000, 2048000};  // complex elements

  const size_t zn = 6 * PL + 3 * AS + 2 * TS;
  se3_zero<<<4096, 256, 0, stream>>>((float4*)ws, zn / 4);

  for (int l = 0; l < 3; ++l)
    se3_emb<<<dim3(24, 8, BB * mls[l]), dim3(32, 4), 0, stream>>>(
        emb[l], x[l], mls[l], kb[l], Qr, Qi, Kr, Ki, Brr, Bii);

  se3_posT<<<(BB * HH * 9 * NP + 255) / 256, 256, 0, stream>>>(
      enc[0], enc[1], enc[2], Qr, Qi, Tr, Ti);

  for (int l = 0; l < 3; ++l)
    se3_scores<<<dim3(NP / 32, NP / 128, BB * HH), dim3(32, 4), 0, stream>>>(
        Qr, Qi, Kr, Ki, Tr, Ti, pos[l], Ar, Ai, mls[l], kb[l], mqd[l], tb[l]);

  se3_softmax<<<BB * HH * NN, 256, 0, stream>>>(Ar, Ai, Aw);

  for (int l = 0; l < 3; ++l)
    se3_av<<<dim3(mqd[l] / 32, NP / 128, BB * HH), dim3(32, 4), 0, stream>>>(
        Brr, Bii, Aw, Vr, Vi, kb[l]);

  for (int l = 0; l < 3; ++l)
    se3_outproj<<<dim3(DIMD / 32, 8, BB * mls[l]), dim3(32, 4), 0, stream>>>(
        outw[l], Vr, Vi, (float2*)d_out, mls[l], kb[l], ob[l]);
}